// SingleDGC_46892452938055
// MI455X (gfx1250) — compile-verified
//
#include <hip/hip_runtime.h>

typedef __attribute__((ext_vector_type(2))) float v2f;
typedef __attribute__((ext_vector_type(8))) float v8f;

#define D_DIM 256
#define EPS_V 1e-5f
#define THR_V 0.5f

__device__ __forceinline__ v8f wmma_f32_k4(v2f a, v2f b, v8f c) {
  // D = A(16x4,f32) * B(4x16,f32) + C(16x16,f32)
  return __builtin_amdgcn_wmma_f32_16x16x4_f32(
      /*neg_a=*/false, a, /*neg_b=*/false, b,
      /*c_mod=*/(short)0, c, /*reuse_a=*/false, /*reuse_b=*/false);
}

// ---------------------------------------------------------------------------
// Kernel 1: xw = X @ W   (N x 256) = (N x 256)(256 x 256), fp32 WMMA 16x16x4.
// Block = 128 threads = 4 waves; each wave computes one 16x16 output tile.
// grid = (4 col-tile groups, N/16 row tiles)
// ---------------------------------------------------------------------------
__global__ void k_gemm(const float* __restrict__ X, const float* __restrict__ W,
                       float* __restrict__ xw) {
  const int wv   = threadIdx.x >> 5;
  const int lane = threadIdx.x & 31;
  const int row0 = blockIdx.y << 4;
  const int col0 = ((blockIdx.x << 2) + wv) << 4;
  const int half = lane >> 4;   // 0: K pair {k,k+1}; 1: K pair {k+2,k+3}
  const int m    = lane & 15;   // A row / B col within tile

  const float* ar = X + (size_t)(row0 + m) * D_DIM;
  const float* wb = W + col0 + m;

  v8f acc = {};
  for (int k = 0; k < D_DIM; k += 4) {
    const int kb = k + 2 * half;
    v2f av = *(const v2f*)(ar + kb);                 // A[m][kb], A[m][kb+1]
    v2f bv;
    bv.x = wb[(size_t)kb * D_DIM];                   // B[kb][col0+m]
    bv.y = wb[(size_t)(kb + 1) * D_DIM];             // B[kb+1][col0+m]
    acc = wmma_f32_k4(av, bv, acc);
  }
  // C/D layout: VGPR v -> row (v + 8*half), col = lane&15
  float* orow = xw + (size_t)(row0 + 8 * half) * D_DIM + (col0 + m);
#pragma unroll
  for (int v = 0; v < 8; ++v) orow[(size_t)v * D_DIM] = acc[v];
}

// ---------------------------------------------------------------------------
// CSR construction: zero -> histogram -> single-block scan -> bucket fill
// ---------------------------------------------------------------------------
__global__ void k_zero(unsigned* __restrict__ deg, int n) {
  int i = blockIdx.x * blockDim.x + threadIdx.x;
  if (i < n) deg[i] = 0u;
}

__global__ void k_count(const int* __restrict__ ei, int E, unsigned* __restrict__ deg) {
  int e = blockIdx.x * blockDim.x + threadIdx.x;
  if (e < E) atomicAdd(&deg[ei[E + e]], 1u);   // row 1 = targets
}

// N must be 16384: 1024 threads x 16 elements, Hillis-Steele block scan.
__global__ void k_scan(const unsigned* __restrict__ deg, unsigned* __restrict__ offs,
                       unsigned* __restrict__ cur, float* __restrict__ dinv) {
  __shared__ unsigned sums[1024];
  const int t = threadIdx.x;
  const int base = t * 16;
  unsigned local[16];
  unsigned run = 0;
#pragma unroll
  for (int i = 0; i < 16; ++i) { local[i] = run; run += deg[base + i]; }
  sums[t] = run;
  __syncthreads();
  for (int o = 1; o < 1024; o <<= 1) {
    unsigned v = (t >= o) ? sums[t - o] : 0u;
    __syncthreads();
    sums[t] += v;
    __syncthreads();
  }
  const unsigned ex = sums[t] - run;   // exclusive prefix of this thread's chunk
#pragma unroll
  for (int i = 0; i < 16; ++i) {
    const unsigned o = ex + local[i];
    offs[base + i] = o;
    cur[base + i]  = o;
    dinv[base + i] = rsqrtf((float)(deg[base + i] + 1u));  // +1 self-loop
  }
}

__global__ void k_fill(const int* __restrict__ ei, int E,
                       unsigned* __restrict__ cur, unsigned* __restrict__ csrc) {
  int e = blockIdx.x * blockDim.x + threadIdx.x;
  if (e < E) {
    const unsigned slot = atomicAdd(&cur[ei[E + e]], 1u);
    csrc[slot] = (unsigned)ei[e];
  }
}

// ---------------------------------------------------------------------------
// Gather: one wave per target node. lane owns 8 contiguous floats (2x float4).
// out[t] = dinv[t]^2 * xw[t] + sum_e dinv[src]*dinv[t] * xw[src] + bias
// ---------------------------------------------------------------------------
__global__ void k_gather(const float* __restrict__ xw, const unsigned* __restrict__ offs,
                         const unsigned* __restrict__ deg, const float* __restrict__ dinv,
                         const unsigned* __restrict__ csrc, const float* __restrict__ bias,
                         float* __restrict__ out) {
  const int wv   = threadIdx.x >> 5;
  const int lane = threadIdx.x & 31;
  const int t    = blockIdx.x * (blockDim.x >> 5) + wv;
  const int e0   = lane << 3;

  const float dt = dinv[t];
  const float4* srow = (const float4*)(xw + (size_t)t * D_DIM + e0);
  const float4 a0 = srow[0], a1 = srow[1];
  const float s = dt * dt;
  float acc[8] = { a0.x * s, a0.y * s, a0.z * s, a0.w * s,
                   a1.x * s, a1.y * s, a1.z * s, a1.w * s };

  const unsigned start = offs[t];
  const unsigned cnt   = deg[t];
  for (unsigned i = 0; i < cnt; ++i) {
    const unsigned src = csrc[start + i];
    const float c = dinv[src] * dt;
    const float4* r = (const float4*)(xw + (size_t)src * D_DIM + e0);
    const float4 r0 = r[0], r1 = r[1];
    acc[0] = fmaf(r0.x, c, acc[0]); acc[1] = fmaf(r0.y, c, acc[1]);
    acc[2] = fmaf(r0.z, c, acc[2]); acc[3] = fmaf(r0.w, c, acc[3]);
    acc[4] = fmaf(r1.x, c, acc[4]); acc[5] = fmaf(r1.y, c, acc[5]);
    acc[6] = fmaf(r1.z, c, acc[6]); acc[7] = fmaf(r1.w, c, acc[7]);
  }

  const float4* bb = (const float4*)(bias + e0);
  const float4 b0 = bb[0], b1 = bb[1];
  float4 o0 = { acc[0] + b0.x, acc[1] + b0.y, acc[2] + b0.z, acc[3] + b0.w };
  float4 o1 = { acc[4] + b1.x, acc[5] + b1.y, acc[6] + b1.z, acc[7] + b1.w };
  float4* op = (float4*)(out + (size_t)t * D_DIM + e0);
  op[0] = o0; op[1] = o1;
}

// ---------------------------------------------------------------------------
// Scores: one block (4 waves) per batch. WMMA Gram (32x32, K=256) -> LDS,
// then per-row (wave32 = 32 cols) shfl_xor min/max, normalize, threshold.
// ---------------------------------------------------------------------------
__global__ void k_scores(const float* __restrict__ Xo, float* __restrict__ mask) {
  __shared__ float g[32][33];
  const int b    = blockIdx.x;
  const float* xb = Xo + (size_t)b * 32 * D_DIM;
  const int wv   = threadIdx.x >> 5;
  const int lane = threadIdx.x & 31;
  const int a0   = (wv >> 1) << 4;
  const int n0   = (wv & 1) << 4;
  const int half = lane >> 4;
  const int idx  = lane & 15;

  const float* ar = xb + (size_t)(a0 + idx) * D_DIM;
  const float* br = xb + (size_t)(n0 + idx) * D_DIM;

  v8f acc = {};
  for (int k = 0; k < D_DIM; k += 4) {
    const int kb = k + 2 * half;
    v2f av = *(const v2f*)(ar + kb);   // A[idx][kb..kb+1]
    v2f bv = *(const v2f*)(br + kb);   // B[kb..kb+1][idx] = Xo[idx][kb..]  (B = Xo^T)
    acc = wmma_f32_k4(av, bv, acc);
  }
#pragma unroll
  for (int v = 0; v < 8; ++v) g[a0 + 8 * half + v][n0 + idx] = acc[v];
  __syncthreads();

  for (int r = wv; r < 32; r += 4) {
    const float sc = g[r][r] - 2.0f * g[r][lane] + g[lane][lane];
    float mn = sc, mx = sc;
#pragma unroll
    for (int o = 16; o >= 1; o >>= 1) {
      mn = fminf(mn, __shfl_xor(mn, o, 32));
      mx = fmaxf(mx, __shfl_xor(mx, o, 32));
    }
    const float nrm = (sc - mn) / (mx - mn + EPS_V);
    mask[(size_t)b * 1024 + r * 32 + lane] = (nrm > THR_V) ? 1.0f : 0.0f;
  }
}

// ---------------------------------------------------------------------------
extern "C" void kernel_launch(void* const* d_in, const int* in_sizes, int n_in,
                              void* d_out, int out_size, void* d_ws, size_t ws_size,
                              hipStream_t stream) {
  const float* X    = (const float*)d_in[0];
  const int*   ei   = (const int*)d_in[1];
  const float* W    = (const float*)d_in[2];
  const float* bias = (const float*)d_in[3];
  const int N = in_sizes[0] / D_DIM;   // 16384
  const int E = in_sizes[1] / 2;       // 1048576

  float*    xw   = (float*)d_ws;                         // N*256 f32 (16 MB)
  unsigned* deg  = (unsigned*)(xw + (size_t)N * D_DIM);  // N
  unsigned* offs = deg + N;                              // N
  unsigned* cur  = offs + N;                             // N
  float*    dinv = (float*)(cur + N);                    // N
  unsigned* csrc = (unsigned*)(dinv + N);                // E

  float* Xo   = (float*)d_out;
  float* mask = Xo + (size_t)N * D_DIM;

  k_gemm  <<<dim3(D_DIM / 64, N / 16), 128, 0, stream>>>(X, W, xw);
  k_zero  <<<(N + 255) / 256, 256, 0, stream>>>(deg, N);
  k_count <<<(E + 255) / 256, 256, 0, stream>>>(ei, E, deg);
  k_scan  <<<1, 1024, 0, stream>>>(deg, offs, cur, dinv);
  k_fill  <<<(E + 255) / 256, 256, 0, stream>>>(ei, E, cur, csrc);
  k_gather<<<N / 8, 256, 0, stream>>>(xw, offs, deg, dinv, csrc, bias, Xo);
  k_scores<<<N / 32, 128, 0, stream>>>(Xo, mask);
}